// DeepLION2_78881369359062
// MI455X (gfx1250) — compile-verified
//
#include <hip/hip_runtime.h>
#include <hip/hip_bf16.h>

typedef float v2f __attribute__((ext_vector_type(2)));
typedef float v8f __attribute__((ext_vector_type(8)));

static constexpr int Bn  = 16;
static constexpr int Nn  = 1024;
static constexpr int Fd  = 15;
static constexpr int Ld  = 24;
static constexpr int Hh  = 4;
static constexpr int HDd = 16;
static constexpr int HIDd = 64;
static constexpr int Cc  = 192;
static constexpr int TK  = 51;
static constexpr int TKP = 52;          // padded
static constexpr float SCALE = 0.25f;   // 1/sqrt(16)

// Output layout (floats): pred[32] | probs[B*H*N*TK] | ctx[B*N*C] | ctx_complete[B*N*C]
static constexpr size_t O_PROBS = 32;
static constexpr size_t O_CTX   = O_PROBS + (size_t)Bn * Hh * Nn * TK;   // 3342368
static constexpr size_t O_CTXC  = O_CTX + (size_t)Bn * Nn * Cc;          // 6488096

#define WMMA_F32(c, a, b) \
  (c) = __builtin_amdgcn_wmma_f32_16x16x4_f32(false, (a), false, (b), (short)0, (c), false, false)

// ---------------------------------------------------------------------------
// Kernel 1: conv branches (k=2..7, 32 ch each) + ReLU + global max pool
// One block per (b,n); 192 threads = 6 branches x 32 out-channels.
// ---------------------------------------------------------------------------
__global__ void conv_feats_kernel(const float* __restrict__ x,
                                  const float* __restrict__ w0, const float* __restrict__ c0,
                                  const float* __restrict__ w1, const float* __restrict__ c1,
                                  const float* __restrict__ w2, const float* __restrict__ c2,
                                  const float* __restrict__ w3, const float* __restrict__ c3,
                                  const float* __restrict__ w4, const float* __restrict__ c4,
                                  const float* __restrict__ w5, const float* __restrict__ c5,
                                  float* __restrict__ feats) {
  __shared__ float xs[Fd * Ld];
  const int row = blockIdx.x;
  const int tid = threadIdx.x;
  const float* xp = x + (long)row * Fd * Ld;
  for (int i = tid; i < Fd * Ld; i += 192) xs[i] = xp[i];
  __syncthreads();

  const int br = tid >> 5;      // branch 0..5
  const int oc = tid & 31;      // out channel
  const int kz = br + 2;
  const float* warr[6] = {w0, w1, w2, w3, w4, w5};
  const float* barr[6] = {c0, c1, c2, c3, c4, c5};
  const float* w = warr[br] + (long)oc * Fd * kz;

  float best = -3.4e38f;
  for (int p = 0; p <= Ld - kz; ++p) {
    float s = 0.f;
    for (int f = 0; f < Fd; ++f) {
      const float* xr = &xs[f * Ld + p];
      const float* wr = &w[f * kz];
      for (int j = 0; j < kz; ++j) s += wr[j] * xr[j];
    }
    best = fmaxf(best, s);
  }
  feats[(long)row * Cc + tid] = fmaxf(best + barr[br][oc], 0.f);
}

// ---------------------------------------------------------------------------
// Kernel 2: Q/K/V projections via f32 WMMA.  [16384 x 192] x [192 x 64]
// grid (1024 mtiles, 3 matrices), block 128; wave w = head tile w.
// Output head-major [B, H, N, HD].
// ---------------------------------------------------------------------------
__global__ void qkv_kernel(const float* __restrict__ feats,
                           const float* __restrict__ qw, const float* __restrict__ qb,
                           const float* __restrict__ kw, const float* __restrict__ kbv,
                           const float* __restrict__ vw, const float* __restrict__ vb,
                           float* __restrict__ qo, float* __restrict__ ko,
                           float* __restrict__ vo) {
  const int mt = blockIdx.x;
  const int which = blockIdx.y;
  const int wv = threadIdx.x >> 5;
  const int lane = threadIdx.x & 31;
  const int lh = lane >> 4, ln = lane & 15;

  const float* W; const float* bias; float* O;
  if (which == 0)      { W = qw; bias = qb;  O = qo; }
  else if (which == 1) { W = kw; bias = kbv; O = ko; }
  else                 { W = vw; bias = vb;  O = vo; }

  const int nt = wv;  // 4 ntiles == 4 heads
  const float* apr = feats + ((long)mt * 16 + ln) * Cc;
  const float* bpr = W + (long)(nt * 16 + ln) * Cc;

  v8f acc = {};
  for (int s = 0; s < 48; ++s) {
    const int kk = 4 * s + 2 * lh;
    v2f a;  a.x = apr[kk];  a.y = apr[kk + 1];
    v2f bm; bm.x = bpr[kk]; bm.y = bpr[kk + 1];
    WMMA_F32(acc, a, bm);
  }
  const float bs = bias[nt * 16 + ln];
#pragma unroll
  for (int i = 0; i < 8; ++i) {
    const int row = mt * 16 + i + 8 * lh;      // global row in [0, B*N)
    const int bidx = row >> 10, n = row & 1023;
    O[(((long)bidx * Hh + nt) * Nn + n) * HDd + ln] = acc[i] + bs;
  }
}

// ---------------------------------------------------------------------------
// Kernel 3: fused attention. One block per (b, qtile). 4 waves.
// Dynamic LDS: a_r[16][1024] | q_s[4][256] | p_s[4][256] | sck[4][16][52] | tk[16][52]
// ---------------------------------------------------------------------------
__global__ void attn_kernel(const float* __restrict__ qg, const float* __restrict__ kg,
                            const float* __restrict__ vg,
                            float* __restrict__ ctxp, float* __restrict__ ctxcp,
                            float* __restrict__ probs_out) {
  extern __shared__ float sm[];
  float* a_r = sm;                    // 16384 floats
  float* q_s = sm + 16384;            // 1024
  float* p_s = sm + 17408;            // 1024
  float* sck = sm + 18432;            // 3328
  int*   tk  = (int*)(sm + 21760);    // 832 ints

  const int blk = blockIdx.x;
  const int b = blk >> 6, qt = blk & 63;
  const int wv = threadIdx.x >> 5;
  const int lane = threadIdx.x & 31;
  const int lh = lane >> 4, ln = lane & 15;

  // Phase 0: stage Q tiles (wave w loads head w)
  {
    const float* qp = qg + (((long)b * Hh + wv) * Nn + qt * 16) * HDd;
    for (int i = lane; i < 256; i += 32) q_s[wv * 256 + i] = qp[i];
  }
  __syncthreads();

  // ---- Phase A: a_r (head-summed raw scores), wave-partitioned over ktiles
  for (int t16 = 0; t16 < 16; ++t16) {
    const int kt = t16 * 4 + wv;
    const int kb = kt * 16;
    v8f asum = {};
    for (int h = 0; h < Hh; ++h) {
      const float* kr = kg + (((long)b * Hh + h) * Nn + kb + ln) * HDd;
      v2f aq[4], bk[4];
#pragma unroll
      for (int s = 0; s < 4; ++s) {
        const int kk = 4 * s + 2 * lh;
        aq[s].x = q_s[h * 256 + ln * 16 + kk];
        aq[s].y = q_s[h * 256 + ln * 16 + kk + 1];
        bk[s].x = kr[kk];
        bk[s].y = kr[kk + 1];
      }
      v8f sc = {};
#pragma unroll
      for (int s = 0; s < 4; ++s) WMMA_F32(sc, aq[s], bk[s]);
#pragma unroll
      for (int i = 0; i < 8; ++i) asum[i] += sc[i];
    }
#pragma unroll
    for (int i = 0; i < 8; ++i) a_r[(i + 8 * lh) * 1024 + kb + ln] = asum[i];
  }
  __syncthreads();

  // ---- Phase B: complete attention (flash softmax), wave w = head w
  {
    const int h = wv;
    const long hb = ((long)b * Hh + h) * Nn;
    v2f aq[4];
#pragma unroll
    for (int s = 0; s < 4; ++s) {
      const int kk = 4 * s + 2 * lh;
      aq[s].x = q_s[h * 256 + ln * 16 + kk];
      aq[s].y = q_s[h * 256 + ln * 16 + kk + 1];
    }
    float m8[8], l8[8];
#pragma unroll
    for (int i = 0; i < 8; ++i) { m8[i] = -3.0e38f; l8[i] = 0.f; }
    v8f acc = {};

    for (int kt = 0; kt < 64; ++kt) {
      const int kb = kt * 16;
      const float* kr = kg + (hb + kb + ln) * HDd;
      if (kt < 63) __builtin_prefetch(kg + (hb + kb + 16 + ln) * HDd, 0, 0);
      v2f bk[4];
#pragma unroll
      for (int s = 0; s < 4; ++s) {
        const int kk = 4 * s + 2 * lh;
        bk[s].x = kr[kk];
        bk[s].y = kr[kk + 1];
      }
      v8f sc = {};
#pragma unroll
      for (int s = 0; s < 4; ++s) WMMA_F32(sc, aq[s], bk[s]);

      // online softmax per query row (row i+8*lh lives in this half-wave)
#pragma unroll
      for (int i = 0; i < 8; ++i) {
        float t = sc[i] * SCALE;
        float rm = t;
        for (int off = 1; off < 16; off <<= 1) rm = fmaxf(rm, __shfl_xor(rm, off, 16));
        const float mn = fmaxf(m8[i], rm);
        const float al = __expf(m8[i] - mn);
        const float p = __expf(t - mn);
        float ps = p;
        for (int off = 1; off < 16; off <<= 1) ps += __shfl_xor(ps, off, 16);
        l8[i] = l8[i] * al + ps;
        m8[i] = mn;
        acc[i] *= al;
        p_s[wv * 256 + (i + 8 * lh) * 16 + ln] = p;   // C-layout -> LDS
      }
      asm volatile("s_wait_dscnt 0x0" ::: "memory");

      // read P in A-layout, V in B-layout; acc += P x V
      v2f ap4[4], bv4[4];
#pragma unroll
      for (int s = 0; s < 4; ++s) {
        const int kk = 4 * s + 2 * lh;
        ap4[s].x = p_s[wv * 256 + ln * 16 + kk];
        ap4[s].y = p_s[wv * 256 + ln * 16 + kk + 1];
        bv4[s].x = vg[(hb + kb + kk) * HDd + ln];
        bv4[s].y = vg[(hb + kb + kk + 1) * HDd + ln];
      }
#pragma unroll
      for (int s = 0; s < 4; ++s) WMMA_F32(acc, ap4[s], bv4[s]);
    }

#pragma unroll
    for (int i = 0; i < 8; ++i) {
      const int r = i + 8 * lh;
      ctxcp[(((long)b * Nn + qt * 16 + r) * Hh + h) * HDd + ln] = acc[i] / l8[i];
    }
  }
  __syncthreads();

  // ---- Top-K (stable, descending) per query row; wave w handles rows w,4+w,8+w,12+w
  for (int rr = 0; rr < 4; ++rr) {
    const int r = wv + rr * 4;
    for (int j = 0; j < TK; ++j) {
      float bvv = -3.4e38f; int bi = 1 << 30;
      for (int c = lane; c < 1024; c += 32) {
        const float v = a_r[r * 1024 + c];
        if (v > bvv || (v == bvv && c < bi)) { bvv = v; bi = c; }
      }
#pragma unroll
      for (int off = 16; off; off >>= 1) {
        const float ov = __shfl_xor(bvv, off, 32);
        const int oi = __shfl_xor(bi, off, 32);
        if (ov > bvv || (ov == bvv && oi < bi)) { bvv = ov; bi = oi; }
      }
      if (lane == 0) { tk[r * TKP + j] = bi; a_r[r * 1024 + bi] = -3.4e38f; }
      asm volatile("s_wait_dscnt 0x0" ::: "memory");
    }
  }
  __syncthreads();

  // ---- Sparse attention, wave w = head w
  {
    const int h = wv;
    const long hb = ((long)b * Hh + h) * Nn;
    // gathered scaled scores
    for (int t = lane; t < 16 * TK; t += 32) {
      const int q = t / TK, j = t - q * TK;
      const int idx = tk[q * TKP + j];
      const float* kr = kg + (hb + idx) * HDd;
      const float* qr = &q_s[h * 256 + q * 16];
      float d = 0.f;
#pragma unroll
      for (int dd = 0; dd < HDd; ++dd) d += qr[dd] * kr[dd];
      sck[h * 16 * TKP + q * TKP + j] = d * SCALE;
    }
    asm volatile("s_wait_dscnt 0x0" ::: "memory");

    // softmax over K dim + write probs
    if (lane < 16) {
      const int q = lane;
      float* srow = &sck[h * 16 * TKP + q * TKP];
      float mx = -3.4e38f;
      for (int j = 0; j < TK; ++j) mx = fmaxf(mx, srow[j]);
      float sum = 0.f;
      for (int j = 0; j < TK; ++j) { const float e = __expf(srow[j] - mx); srow[j] = e; sum += e; }
      const float inv = 1.f / sum;
      float* po = probs_out + (((long)b * Hh + h) * Nn + qt * 16 + q) * TK;
      for (int j = 0; j < TK; ++j) { const float p = srow[j] * inv; srow[j] = p; po[j] = p; }
    }
    asm volatile("s_wait_dscnt 0x0" ::: "memory");

    // ctx = probs @ gathered V ; lanes = 2 queries x 16 dims per pass
    for (int qq = 0; qq < 8; ++qq) {
      const int q = qq * 2 + lh;
      const int d = ln;
      float a = 0.f;
      for (int j = 0; j < TK; ++j) {
        const float p = sck[h * 16 * TKP + q * TKP + j];
        const int idx = tk[q * TKP + j];
        a += p * vg[(hb + idx) * HDd + d];
      }
      ctxp[(((long)b * Nn + qt * 16 + q) * Hh + h) * HDd + d] = a;
    }
  }
}

// ---------------------------------------------------------------------------
// Kernel 4: output projection via WMMA: [16384 x 64] x [64 x 192] + attn_b
// grid (1024 mtiles, 2 branches sparse/complete), block 128.
// ---------------------------------------------------------------------------
__global__ void outproj_kernel(const float* __restrict__ ctxp, const float* __restrict__ ctxcp,
                               const float* __restrict__ aw, const float* __restrict__ ab,
                               float* __restrict__ out) {
  const int mt = blockIdx.x;
  const int which = blockIdx.y;
  const int wv = threadIdx.x >> 5;
  const int lane = threadIdx.x & 31;
  const int lh = lane >> 4, ln = lane & 15;

  const float* src = which ? ctxcp : ctxp;
  float* dst = out + (which ? O_CTXC : O_CTX);
  const float* apr = src + ((long)mt * 16 + ln) * HIDd;

  for (int rnt = 0; rnt < 3; ++rnt) {
    const int nt = wv + rnt * 4;   // 12 ntiles across 4 waves
    const float* bpr = aw + (long)(nt * 16 + ln) * HIDd;
    v8f acc = {};
    for (int s = 0; s < 16; ++s) {
      const int kk = 4 * s + 2 * lh;
      v2f a;  a.x = apr[kk];  a.y = apr[kk + 1];
      v2f bm; bm.x = bpr[kk]; bm.y = bpr[kk + 1];
      WMMA_F32(acc, a, bm);
    }
    const float bs = ab[nt * 16 + ln];
#pragma unroll
    for (int i = 0; i < 8; ++i) {
      const long row = (long)mt * 16 + i + 8 * lh;
      dst[row * Cc + nt * 16 + ln] = acc[i] + bs;
    }
  }
}

// ---------------------------------------------------------------------------
// Kernel 5: MIL pooling: pred[b,o] = sum_n(ctx[b,n,:] @ mil_w[o,:]) / N + mil_b[o]
// Deterministic fixed-order LDS tree reduction. One block per batch.
// ---------------------------------------------------------------------------
__global__ void pred_kernel(const float* __restrict__ ctx_out,
                            const float* __restrict__ milw, const float* __restrict__ milb,
                            float* __restrict__ out) {
  __shared__ float red[512];
  const int b = blockIdx.x, t = threadIdx.x;
  float s0 = 0.f, s1 = 0.f;
  for (int n = t; n < Nn; n += 256) {
    const float* r = ctx_out + ((long)b * Nn + n) * Cc;
    for (int c = 0; c < Cc; ++c) {
      const float v = r[c];
      s0 += v * milw[c];
      s1 += v * milw[Cc + c];
    }
  }
  red[t] = s0; red[256 + t] = s1;
  __syncthreads();
  for (int off = 128; off; off >>= 1) {
    if (t < off) { red[t] += red[t + off]; red[256 + t] += red[256 + t + off]; }
    __syncthreads();
  }
  if (t == 0) {
    out[b * 2 + 0] = red[0] * (1.f / Nn) + milb[0];
    out[b * 2 + 1] = red[256] * (1.f / Nn) + milb[1];
  }
}

// ---------------------------------------------------------------------------
extern "C" void kernel_launch(void* const* d_in, const int* in_sizes, int n_in,
                              void* d_out, int out_size, void* d_ws, size_t ws_size,
                              hipStream_t stream) {
  (void)in_sizes; (void)n_in; (void)out_size; (void)ws_size;
  const float* x = (const float*)d_in[0];
  const float* cw[6]; const float* cb[6];
  for (int i = 0; i < 6; ++i) {
    cw[i] = (const float*)d_in[1 + 2 * i];
    cb[i] = (const float*)d_in[2 + 2 * i];
  }
  const float* qw = (const float*)d_in[13];
  const float* qb = (const float*)d_in[14];
  const float* kw = (const float*)d_in[15];
  const float* kb = (const float*)d_in[16];
  const float* vw = (const float*)d_in[17];
  const float* vb = (const float*)d_in[18];
  const float* aw = (const float*)d_in[19];
  const float* ab = (const float*)d_in[20];
  const float* mw = (const float*)d_in[21];
  const float* mb = (const float*)d_in[22];

  float* out = (float*)d_out;
  float* ws = (float*)d_ws;
  float* feats = ws;                              // 16384*192 = 3145728
  float* qg    = feats + (size_t)Bn * Nn * Cc;    // 1048576 each
  float* kg    = qg + (size_t)Bn * Hh * Nn * HDd;
  float* vg    = kg + (size_t)Bn * Hh * Nn * HDd;
  float* ctxp  = vg + (size_t)Bn * Hh * Nn * HDd;
  float* ctxcp = ctxp + (size_t)Bn * Nn * HIDd;

  conv_feats_kernel<<<Bn * Nn, 192, 0, stream>>>(x, cw[0], cb[0], cw[1], cb[1], cw[2], cb[2],
                                                 cw[3], cb[3], cw[4], cb[4], cw[5], cb[5], feats);
  qkv_kernel<<<dim3(1024, 3), 128, 0, stream>>>(feats, qw, qb, kw, kb, vw, vb, qg, kg, vg);
  attn_kernel<<<Bn * 64, 128, 90368, stream>>>(qg, kg, vg, ctxp, ctxcp, out + O_PROBS);
  outproj_kernel<<<dim3(1024, 2), 128, 0, stream>>>(ctxp, ctxcp, aw, ab, out);
  pred_kernel<<<Bn, 256, 0, stream>>>(out + O_CTX, mw, mb, out);
}